// LocalAttention_21560735826482
// MI455X (gfx1250) — compile-verified
//
#include <hip/hip_runtime.h>
#include <cfloat>

typedef __attribute__((ext_vector_type(2))) float v2f;
typedef __attribute__((ext_vector_type(8))) float v8f;

#define SK_STRIDE 68    // floats; 68*4=272 -> 8B aligned rows, bank pattern 4*row+c (conflict free)
#define SVT_STRIDE 260  // floats; 260*4=1040 -> 8B aligned, bank pattern 4*col+j (conflict free)
#define SP_STRIDE 34    // floats; 34*4=136 -> 8B aligned

__device__ __forceinline__ v8f wmma_f32_16x16x4(v2f a, v2f b, v8f c) {
  // (neg_a, A, neg_b, B, c_mod, C, reuse_a, reuse_b)
  return __builtin_amdgcn_wmma_f32_16x16x4_f32(false, a, false, b, (short)0, c, false, false);
}

__global__ __launch_bounds__(256) void local_attn_rope_kernel(
    const float* __restrict__ q, const float* __restrict__ k,
    const float* __restrict__ v, float* __restrict__ out)
{
  __shared__ float sK [256 * SK_STRIDE];   // RoPE'd keys, row-major, local frame j=0..255
  __shared__ float sVT[ 64 * SVT_STRIDE];  // values transposed: sVT[d_col][j]
  __shared__ float sP [8 * 16 * SP_STRIDE];// per-wave P chunk (16 rows x 32 cols)

  const int wi   = blockIdx.x;   // window 0..31
  const int bb   = blockIdx.y;   // batch 0..31
  const int tid  = threadIdx.x;
  const int wv   = tid >> 5;     // wave 0..7: owns query rows 16*wv..16*wv+15
  const int lane = tid & 31;
  const int l16  = lane & 15;
  const int hi   = lane >> 4;

  const float* kb = k + (size_t)bb * 4096 * 64;
  const float* vb = v + (size_t)bb * 4096 * 64;
  const int jbase = wi * 128 - 128;              // global row of local key j=0

  const float LOG2_IF = -0.41524101186092029f;   // -log2(10000)/32

  // ---- stage K into LDS with RoPE (keys use t = local j = 0..255) ----
  #pragma unroll
  for (int it = 0; it < 32; ++it) {
    int p   = tid + 256 * it;     // 0..8191 = (row, c) pairs, c = 0..31
    int row = p >> 5;
    int c   = p & 31;
    int grow = jbase + row;
    float x1 = 0.f, x2 = 0.f;
    if (grow >= 0) {
      const float* kr = kb + (size_t)grow * 64;
      x1 = kr[c]; x2 = kr[c + 32];
    }
    float ang = (float)row * exp2f(LOG2_IF * (float)c);
    float sn, cs; __sincosf(ang, &sn, &cs);
    sK[row * SK_STRIDE + c]      = x1 * cs - x2 * sn;
    sK[row * SK_STRIDE + c + 32] = x2 * cs + x1 * sn;
  }

  // ---- stage V transposed into LDS ----
  #pragma unroll
  for (int it = 0; it < 16; ++it) {
    int p   = tid + 256 * it;     // float4 index
    int row = p >> 4;
    int c4  = (p & 15) * 4;
    int grow = jbase + row;
    float4 val = make_float4(0.f, 0.f, 0.f, 0.f);
    if (grow >= 0) val = *(const float4*)(vb + (size_t)grow * 64 + c4);
    sVT[(c4 + 0) * SVT_STRIDE + row] = val.x;
    sVT[(c4 + 1) * SVT_STRIDE + row] = val.y;
    sVT[(c4 + 2) * SVT_STRIDE + row] = val.z;
    sVT[(c4 + 3) * SVT_STRIDE + row] = val.w;
  }

  // ---- load Q A-fragments from global, RoPE + scale in registers ----
  // A 16x4 f32 layout: lane holds row = l16, k = 4t + 2*hi + {0,1} per step t.
  // rotate_half partner k<->k+32 keeps (k>>1)&1 -> same lane, regs t and t+8.
  const int iq = 16 * wv + l16;
  const float* qrow = q + ((size_t)bb * 4096 + (size_t)wi * 128 + iq) * 64;
  v2f qf[16];
  #pragma unroll
  for (int t = 0; t < 16; ++t)
    qf[t] = *(const v2f*)(qrow + 4 * t + 2 * hi);
  {
    const float scale = 0.125f;        // 64^-0.5, folded into cos/sin
    float pos = (float)(128 + iq);     // q_freqs = freqs[128:]
    #pragma unroll
    for (int t = 0; t < 8; ++t) {
      #pragma unroll
      for (int i = 0; i < 2; ++i) {
        int kk = 4 * t + 2 * hi + i;   // < 32
        float ang = pos * exp2f(LOG2_IF * (float)kk);
        float sn, cs; __sincosf(ang, &sn, &cs);
        cs *= scale; sn *= scale;
        float lo = qf[t][i], hh = qf[t + 8][i];
        qf[t][i]     = lo * cs - hh * sn;
        qf[t + 8][i] = hh * cs + lo * sn;
      }
    }
  }

  __syncthreads();

  // ---- S = Qr * Kr^T : 16 tiles of 16x16, K-dim 64 in steps of 4 ----
  v8f S[16];
  #pragma unroll
  for (int jt = 0; jt < 16; ++jt) {
    v8f acc = {0.f,0.f,0.f,0.f,0.f,0.f,0.f,0.f};
    const float* kr = &sK[(16 * jt + l16) * SK_STRIDE + 2 * hi];
    #pragma unroll
    for (int t = 0; t < 16; ++t) {
      v2f bf = *(const v2f*)(kr + 4 * t);          // ds_load_b64
      acc = wmma_f32_16x16x4(qf[t], bf, acc);
    }
    S[jt] = acc;
  }

  // ---- masking (C layout: row = vv + 8*hi, col j = 16*jt + l16) ----
  const float NEGMAX = -3.402823466e38f;
  #pragma unroll
  for (int jt = 0; jt < 16; ++jt) {
    int j = 16 * jt + l16;
    bool padj = (wi == 0) && (j < 128);
    #pragma unroll
    for (int vv = 0; vv < 8; ++vv) {
      int iqc = 16 * wv + vv + 8 * hi;
      if (padj || (j > iqc + 128)) S[jt][vv] = NEGMAX;
    }
  }

  // ---- row softmax (reduce across 16 lanes of each half-wave) ----
  float rinv[8];
  #pragma unroll
  for (int vv = 0; vv < 8; ++vv) {
    float m = S[0][vv];
    #pragma unroll
    for (int jt = 1; jt < 16; ++jt) m = fmaxf(m, S[jt][vv]);
    m = fmaxf(m, __shfl_xor(m, 1, 16));
    m = fmaxf(m, __shfl_xor(m, 2, 16));
    m = fmaxf(m, __shfl_xor(m, 4, 16));
    m = fmaxf(m, __shfl_xor(m, 8, 16));
    float s = 0.f;
    #pragma unroll
    for (int jt = 0; jt < 16; ++jt) {
      float e = __expf(S[jt][vv] - m);
      S[jt][vv] = e;
      s += e;
    }
    s += __shfl_xor(s, 1, 16);
    s += __shfl_xor(s, 2, 16);
    s += __shfl_xor(s, 4, 16);
    s += __shfl_xor(s, 8, 16);
    rinv[vv] = 1.0f / s;
  }

  // ---- O = P * V : P redistributed C->A layout via per-wave LDS chunks ----
  v8f O[4];
  #pragma unroll
  for (int n = 0; n < 4; ++n) O[n] = (v8f){0.f,0.f,0.f,0.f,0.f,0.f,0.f,0.f};

  float* myP = &sP[wv * 16 * SP_STRIDE];
  #pragma unroll
  for (int ch = 0; ch < 8; ++ch) {     // 32 key-columns per chunk
    #pragma unroll
    for (int u = 0; u < 2; ++u) {
      v8f Pv = S[2 * ch + u];
      #pragma unroll
      for (int vv = 0; vv < 8; ++vv)
        myP[(vv + 8 * hi) * SP_STRIDE + 16 * u + l16] = Pv[vv];
    }
    asm volatile("s_wait_dscnt 0" ::: "memory");   // wave-private buffer: wait, no barrier
    #pragma unroll
    for (int t = 0; t < 8; ++t) {
      v2f af = *(const v2f*)(&myP[l16 * SP_STRIDE + 4 * t + 2 * hi]);
      int jb = 32 * ch + 4 * t + 2 * hi;
      #pragma unroll
      for (int n = 0; n < 4; ++n) {
        v2f bf = *(const v2f*)(&sVT[(16 * n + l16) * SVT_STRIDE + jb]); // ds_load_b64
        O[n] = wmma_f32_16x16x4(af, bf, O[n]);
      }
    }
  }

  // ---- normalize + store (C layout: row = 16*wv + vv + 8*hi, col = 16*n + l16) ----
  float* ob = out + ((size_t)bb * 4096 + (size_t)wi * 128 + 16 * wv + 8 * hi) * 64 + l16;
  #pragma unroll
  for (int n = 0; n < 4; ++n) {
    #pragma unroll
    for (int vv = 0; vv < 8; ++vv)
      ob[(size_t)vv * 64 + 16 * n] = O[n][vv] * rinv[vv];
  }
}

extern "C" void kernel_launch(void* const* d_in, const int* in_sizes, int n_in,
                              void* d_out, int out_size, void* d_ws, size_t ws_size,
                              hipStream_t stream) {
  const float* q = (const float*)d_in[0];
  const float* k = (const float*)d_in[1];
  const float* v = (const float*)d_in[2];
  float* out = (float*)d_out;
  dim3 grid(32, 32);   // (window, batch)
  local_attn_rope_kernel<<<grid, dim3(256), 0, stream>>>(q, k, v, out);
}